// Net_45234595561883
// MI455X (gfx1250) — compile-verified
//
#include <hip/hip_runtime.h>
#include <hip/hip_bf16.h>
#include <math.h>

// Problem constants (match reference)
#define KC    128
#define EXERC 16384
#define NC    (KC + EXERC)       // 16512
#define BC    4096
#define EC    400000
#define TC    (BC * 50)          // 204800
#define H1    512
#define H2    216

typedef __attribute__((ext_vector_type(2))) float v2f;
typedef __attribute__((ext_vector_type(8))) float v8f;

__device__ __forceinline__ float sigmoidf(float x) { return 1.0f / (1.0f + expf(-x)); }

// ---------------- degree / norm ----------------
__global__ void deg_kernel(const int* __restrict__ src, const int* __restrict__ dst,
                           float* __restrict__ deg, int E, int N) {
    int e = blockIdx.x * blockDim.x + threadIdx.x;
    if (e >= E) return;
    atomicAdd(&deg[src[e]], 1.0f);
    atomicAdd(&deg[N + dst[e]], 1.0f);
}

__global__ void norm_kernel(const int* __restrict__ src, const int* __restrict__ dst,
                            const float* __restrict__ deg, float* __restrict__ norm,
                            int E, int N) {
    int e = blockIdx.x * blockDim.x + threadIdx.x;
    if (e >= E) return;
    float dn = fmaxf(deg[src[e]], 1.0f) * fmaxf(deg[N + dst[e]], 1.0f);
    norm[e] = rsqrtf(dn);
}

// ---------------- edge gather/scatter: agg[dst] += h[src] * norm ----------------
// one wave per edge; lane moves float4 (32 lanes * 4 = 128 floats)
__global__ void edge_scatter(const int* __restrict__ src, const int* __restrict__ dst,
                             const float* __restrict__ norm, const float* __restrict__ h,
                             float* __restrict__ agg, int E) {
    int e = blockIdx.x * (blockDim.x >> 5) + (threadIdx.x >> 5);
    if (e >= E) return;
    int lane = threadIdx.x & 31;
    int s = src[e], d = dst[e];
    float w = norm[e];
    float4 v = *(const float4*)(h + (size_t)s * KC + lane * 4);
    float* ap = agg + (size_t)d * KC + lane * 4;
    atomicAdd(ap + 0, v.x * w);
    atomicAdd(ap + 1, v.y * w);
    atomicAdd(ap + 2, v.z * w);
    atomicAdd(ap + 3, v.w * w);
}

// ---------------- nh init with summed biases ----------------
__global__ void nh_bias_init(const float* __restrict__ gb, float* __restrict__ nh,
                             int l, size_t NK) {
    size_t i = (size_t)blockIdx.x * blockDim.x + threadIdx.x;
    if (i >= NK) return;
    int k = (int)(i & (KC - 1));
    nh[i] = gb[(0 * 4 + l) * KC + k] + gb[(1 * 4 + l) * KC + k] + gb[(2 * 4 + l) * KC + k];
}

// ---------------- embedding-stack accumulators ----------------
__global__ void accum_kernel(const float* __restrict__ h, float* __restrict__ full,
                             float* __restrict__ disc, int addDisc, size_t NK) {
    size_t i = (size_t)blockIdx.x * blockDim.x + threadIdx.x;
    if (i >= NK) return;
    float v = h[i];
    full[i] += v;
    if (addDisc) disc[i] += v;
}

__global__ void scale_kernel(float* __restrict__ full, float* __restrict__ disc, size_t NK) {
    size_t i = (size_t)blockIdx.x * blockDim.x + threadIdx.x;
    if (i >= NK) return;
    full[i] *= 0.2f;   // mean over 5 embeddings
    disc[i] *= 0.5f;   // mean over last 2
}

// ---------------- fp32 WMMA GEMM ----------------
// C(M x Nb) = [C +] A(M x Kd) * B(Kd x Nb) [+ bias][sigmoid]
// One wave (32 threads) per (16*MT) x 16 output panel: each B float2 load feeds MT WMMAs.
// Out-of-range columns are handled by clamping the load address (branchless); clamped
// lanes produce garbage that is never stored, and each GEMM column is independent.
template<int MT, bool ACCUM, bool BIAS, bool SIG>
__global__ void gemm_wmma_f32(const float* __restrict__ A, const float* __restrict__ Bm,
                              const float* __restrict__ bias, float* __restrict__ C,
                              int Kd, int Nb, int lda, int ldb, int ldc) {
    int m0 = blockIdx.x * (16 * MT);
    int n0 = blockIdx.y * 16;
    int lane = threadIdx.x & 31;
    int half = lane >> 4;     // 0: lanes 0-15, 1: lanes 16-31
    int lr   = lane & 15;
    int col  = n0 + lr;
    bool colOK = col < Nb;
    int colc = colOK ? col : (Nb - 1);   // clamped address, no divergent branch

    v8f acc[MT];
    #pragma unroll
    for (int t = 0; t < MT; ++t) {
        if (ACCUM) {
            #pragma unroll
            for (int r = 0; r < 8; ++r)
                acc[t][r] = C[(size_t)(m0 + t * 16 + r + 8 * half) * ldc + colc];
        } else {
            acc[t] = (v8f){};
        }
    }

    // strength-reduced pointers
    const float* bp  = Bm + (size_t)(2 * half) * ldb + colc;  // rows 2*half, 2*half+1 (+4 per step)
    const float* ap0 = A + (size_t)(m0 + lr) * lda + 2 * half;

    for (int k0 = 0; k0 < Kd; k0 += 4) {
        v2f b;
        b[0] = bp[0];
        b[1] = bp[(size_t)ldb];
        bp += (size_t)4 * ldb;
        #pragma unroll
        for (int t = 0; t < MT; ++t) {
            const float* ap = ap0 + (size_t)(t * 16) * lda + k0;
            v2f a; a[0] = ap[0]; a[1] = ap[1];    // contiguous pair -> b64 load
            acc[t] = __builtin_amdgcn_wmma_f32_16x16x4_f32(false, a, false, b, (short)0,
                                                           acc[t], false, false);
        }
    }

    if (colOK) {
        float bv = BIAS ? bias[col] : 0.0f;
        #pragma unroll
        for (int t = 0; t < MT; ++t) {
            #pragma unroll
            for (int r = 0; r < 8; ++r) {
                float v = acc[t][r];
                if (BIAS) v += bv;
                if (SIG)  v = sigmoidf(v);
                C[(size_t)(m0 + t * 16 + r + 8 * half) * ldc + col] = v;
            }
        }
    }
}

// ---------------- history segment sum (counts + row scatter) ----------------
__global__ void cnt_kernel(const int* __restrict__ hseg, float* __restrict__ cnt, int T) {
    int t = blockIdx.x * blockDim.x + threadIdx.x;
    if (t >= T) return;
    atomicAdd(&cnt[hseg[t]], 1.0f);
}

__global__ void hist_scatter(const float* __restrict__ disc, const int* __restrict__ hid,
                             const int* __restrict__ hseg, float* __restrict__ ssum, int T) {
    int t = blockIdx.x * (blockDim.x >> 5) + (threadIdx.x >> 5);
    if (t >= T) return;
    int lane = threadIdx.x & 31;
    int row = KC + hid[t];                 // stu_exe = disc_full[K:]
    int seg = hseg[t];
    float4 v = *(const float4*)(disc + (size_t)row * KC + lane * 4);
    float* ap = ssum + (size_t)seg * KC + lane * 4;
    atomicAdd(ap + 0, v.x);
    atomicAdd(ap + 1, v.y);
    atomicAdd(ap + 2, v.z);
    atomicAdd(ap + 3, v.w);
}

// ---------------- concept projections: cv[k] = sum_j full[k,j] * W[K+j] ----------------
__global__ void cvec_kernel(const float* __restrict__ full, const float* __restrict__ fsW,
                            const float* __restrict__ feW, float* __restrict__ cvs,
                            float* __restrict__ cve) {
    int k = threadIdx.x;   // 128 threads, 1 block
    float ss = 0.0f, se = 0.0f;
    const float* row = full + (size_t)k * KC;
    for (int j = 0; j < KC; ++j) {
        float v = row[j];
        ss += v * fsW[KC + j];
        se += v * feW[KC + j];
    }
    cvs[k] = ss;
    cve[k] = se;
}

// ---------------- build MLP input: x0 = e_disc * (prof - diff) * kn ----------------
__global__ void build_x(const float* __restrict__ full, const float* __restrict__ ssum,
                        const float* __restrict__ cnt, const float* __restrict__ fsW,
                        const float* __restrict__ fsb, const float* __restrict__ feW,
                        const float* __restrict__ feb, const float* __restrict__ disc_emb,
                        const int* __restrict__ exer_id, const float* __restrict__ kn,
                        const float* __restrict__ cvs, const float* __restrict__ cve,
                        float* __restrict__ x0) {
    int b = blockIdx.x;
    int k = threadIdx.x;   // 128 threads
    __shared__ float red[KC];

    float c = fmaxf(cnt[b], 1.0f);
    float stu_k = ssum[(size_t)b * KC + k] / c;

    red[k] = stu_k * fsW[k];
    __syncthreads();
    for (int s = 64; s > 0; s >>= 1) { if (k < s) red[k] += red[k + s]; __syncthreads(); }
    float svec = red[0];
    __syncthreads();

    int eid = exer_id[b];
    float ex_k = full[(size_t)(KC + eid) * KC + k];   // exer_ent = full[K:]
    red[k] = ex_k * feW[k];
    __syncthreads();
    for (int s = 64; s > 0; s >>= 1) { if (k < s) red[k] += red[k + s]; __syncthreads(); }
    float evec = red[0];

    float prof = sigmoidf(svec + cvs[k] + fsb[0]);
    float diff = sigmoidf(evec + cve[k] + feb[0]);
    float ed = sigmoidf(disc_emb[eid]) * 10.0f;
    x0[(size_t)b * KC + k] = ed * (prof - diff) * kn[(size_t)b * KC + k];
}

// ---------------- final head: out[b] = sigmoid(x2[b,:] . W3 + b3) ----------------
__global__ void head_kernel(const float* __restrict__ x2, const float* __restrict__ W3,
                            const float* __restrict__ b3, float* __restrict__ out, int B) {
    int b = blockIdx.x * blockDim.x + threadIdx.x;
    if (b >= B) return;
    float s = b3[0];
    const float* row = x2 + (size_t)b * H2;
    for (int k = 0; k < H2; ++k) s += row[k] * W3[k];
    out[b] = sigmoidf(s);
}

// =====================================================================================
extern "C" void kernel_launch(void* const* d_in, const int* in_sizes, int n_in,
                              void* d_out, int out_size, void* d_ws, size_t ws_size,
                              hipStream_t stream) {
    (void)in_sizes; (void)n_in; (void)out_size; (void)ws_size;
    const size_t NK = (size_t)NC * KC;

    const float* entity   = (const float*)d_in[0];
    const float* gW       = (const float*)d_in[1];
    const float* gb       = (const float*)d_in[2];
    const float* fsW      = (const float*)d_in[3];
    const float* fsb      = (const float*)d_in[4];
    const float* feW      = (const float*)d_in[5];
    const float* feb      = (const float*)d_in[6];
    const float* disc_emb = (const float*)d_in[7];
    const float* W1       = (const float*)d_in[8];
    const float* b1       = (const float*)d_in[9];
    const float* W2       = (const float*)d_in[10];
    const float* b2       = (const float*)d_in[11];
    const float* W3       = (const float*)d_in[12];
    const float* b3       = (const float*)d_in[13];
    const int* srcs[3] = {(const int*)d_in[14], (const int*)d_in[16], (const int*)d_in[18]};
    const int* dsts[3] = {(const int*)d_in[15], (const int*)d_in[17], (const int*)d_in[19]};
    const int*   exer_id  = (const int*)d_in[21];
    const float* kn       = (const float*)d_in[22];
    const int*   hid      = (const int*)d_in[23];
    const int*   hseg     = (const int*)d_in[24];
    float* out = (float*)d_out;

    // workspace layout (floats); total ~15.9M floats (~64 MB)
    float* ws   = (float*)d_ws;
    size_t off  = 0;
    float* norm = ws + off; off += 3 * (size_t)EC;
    float* deg  = ws + off; off += 2 * (size_t)NC;
    float* hA   = ws + off; off += NK;
    float* hB   = ws + off; off += NK;
    float* agg  = ws + off; off += NK;
    float* full = ws + off; off += NK;
    float* disc = ws + off; off += NK;
    float* ssum = ws + off; off += (size_t)BC * KC;
    float* cnt  = ws + off; off += BC;
    float* cvs  = ws + off; off += KC;
    float* cve  = ws + off; off += KC;
    float* x0   = ws + off; off += (size_t)BC * KC;
    float* x1   = ws + off; off += (size_t)BC * H1;
    float* x2   = ws + off; off += (size_t)BC * H2;

    const int eb = (EC + 255) / 256;
    const int nkb = (int)((NK + 255) / 256);

    // 1) per-graph edge norms
    for (int g = 0; g < 3; ++g) {
        hipMemsetAsync(deg, 0, 2 * (size_t)NC * sizeof(float), stream);
        deg_kernel<<<eb, 256, 0, stream>>>(srcs[g], dsts[g], deg, EC, NC);
        norm_kernel<<<eb, 256, 0, stream>>>(srcs[g], dsts[g], deg, norm + (size_t)g * EC, EC, NC);
    }

    // 2) GCN layers
    hipMemcpyAsync(hA, entity, NK * sizeof(float), hipMemcpyDeviceToDevice, stream);
    hipMemcpyAsync(full, entity, NK * sizeof(float), hipMemcpyDeviceToDevice, stream);
    hipMemsetAsync(disc, 0, NK * sizeof(float), stream);

    float* h = hA;
    float* nh = hB;
    for (int l = 0; l < 4; ++l) {
        nh_bias_init<<<nkb, 256, 0, stream>>>(gb, nh, l, NK);
        for (int g = 0; g < 3; ++g) {
            hipMemsetAsync(agg, 0, NK * sizeof(float), stream);
            edge_scatter<<<(EC + 7) / 8, 256, 0, stream>>>(srcs[g], dsts[g],
                                                           norm + (size_t)g * EC, h, agg, EC);
            // nh += agg @ gcn_W[g,l]   (16512 = 258 * 64 rows)
            gemm_wmma_f32<4, true, false, false>
                <<<dim3(NC / 64, KC / 16), 32, 0, stream>>>(
                    agg, gW + (size_t)(g * 4 + l) * KC * KC, nullptr, nh,
                    KC, KC, KC, KC, KC);
        }
        accum_kernel<<<nkb, 256, 0, stream>>>(nh, full, disc, (l >= 2) ? 1 : 0, NK);
        float* t = h; h = nh; nh = t;
    }
    scale_kernel<<<nkb, 256, 0, stream>>>(full, disc, NK);

    // 3) student history mean
    hipMemsetAsync(ssum, 0, (size_t)BC * KC * sizeof(float), stream);
    hipMemsetAsync(cnt, 0, (size_t)BC * sizeof(float), stream);
    cnt_kernel<<<(TC + 255) / 256, 256, 0, stream>>>(hseg, cnt, TC);
    hist_scatter<<<(TC + 7) / 8, 256, 0, stream>>>(disc, hid, hseg, ssum, TC);

    // 4) rank-1 projections + MLP input
    cvec_kernel<<<1, KC, 0, stream>>>(full, fsW, feW, cvs, cve);
    build_x<<<BC, KC, 0, stream>>>(full, ssum, cnt, fsW, fsb, feW, feb,
                                   disc_emb, exer_id, kn, cvs, cve, x0);

    // 5) MLP with fp32 WMMA (4096 = 64 * 64 rows)
    gemm_wmma_f32<4, false, true, true>
        <<<dim3(BC / 64, H1 / 16), 32, 0, stream>>>(x0, W1, b1, x1, KC, H1, KC, H1, H1);
    gemm_wmma_f32<4, false, true, true>
        <<<dim3(BC / 64, (H2 + 15) / 16), 32, 0, stream>>>(x1, W2, b2, x2, H1, H2, H1, H2, H2);
    head_kernel<<<(BC + 255) / 256, 256, 0, stream>>>(x2, W3, b3, out, BC);
}